// VarianceBasedBinLoss_5849745457960
// MI455X (gfx1250) — compile-verified
//
#include <hip/hip_runtime.h>

typedef float v2f __attribute__((ext_vector_type(2)));
typedef float v8f __attribute__((ext_vector_type(8)));

// Only meaningful on the device pass; the host pass never declares amdgcn builtins.
#if defined(__HIP_DEVICE_COMPILE__) && !__has_builtin(__builtin_amdgcn_wmma_f32_16x16x4_f32)
#error "device toolchain lacks __builtin_amdgcn_wmma_f32_16x16x4_f32"
#endif

#define NFEAT 32
#define NBIN  16
#define NCLS  10
#define FKDIM 512            // NFEAT*NBIN
#define CHUNK 256            // n's per block (multiple of 4)
#define WSFLOATS (2 * NFEAT * 8 * 32)   // 2 GEMMs x 32 f-tiles x 8 VGPRs x 32 lanes

static __device__ __forceinline__ v8f wmma_f32x4(v2f a, v2f b, v8f c) {
#if defined(__HIP_DEVICE_COMPILE__)
    return __builtin_amdgcn_wmma_f32_16x16x4_f32(false, a, false, b, (short)0, c, false, false);
#else
    return c;
#endif
}

// DPP16 row_ror:N move (ctrl = 0x120+N): rotate within each 16-lane row.
// Pure VALU lane exchange -- no LDS traffic, no dscnt waits.
template <int CTRL>
static __device__ __forceinline__ float dpp_ror_f32(float x) {
#if defined(__HIP_DEVICE_COMPILE__)
    int r = __builtin_amdgcn_update_dpp(0, __builtin_bit_cast(int, x),
                                        CTRL, 0xF, 0xF, true);
    return __builtin_bit_cast(float, r);
#else
    return x;
#endif
}

// Full 16-lane-row reductions via rotate butterfly (result in every lane).
static __device__ __forceinline__ float row16_max(float x) {
    x = fmaxf(x, dpp_ror_f32<0x121>(x));   // ror 1
    x = fmaxf(x, dpp_ror_f32<0x122>(x));   // ror 2
    x = fmaxf(x, dpp_ror_f32<0x124>(x));   // ror 4
    x = fmaxf(x, dpp_ror_f32<0x128>(x));   // ror 8
    return x;
}
static __device__ __forceinline__ float row16_sum(float x) {
    x += dpp_ror_f32<0x121>(x);
    x += dpp_ror_f32<0x122>(x);
    x += dpp_ror_f32<0x124>(x);
    x += dpp_ror_f32<0x128>(x);
    return x;
}

__global__ void vb_zero(float* __restrict__ ws, int n) {
    int i = blockIdx.x * blockDim.x + threadIdx.x;
    if (i < n) ws[i] = 0.0f;
}

// GEMM kernel: D_soft[fk, col] = sum_n m[n,fk] * B[n,col]
//              D_hard[fk, col] = sum_n onehot[n,fk] * B[n,col]
// B cols: 0..9 = y[n,c], 10 = 1.0, 11 = ||y_n||^2, 12..15 = 0
__global__ __launch_bounds__(256) void vb_gemm(const float* __restrict__ mem,
                                               const float* __restrict__ tp,
                                               float* __restrict__ ws, int N) {
    const int tid   = threadIdx.x;
    const int wave  = tid >> 5;          // 0..7
    const int lane  = tid & 31;
    const int col   = lane & 15;         // M-row of A (=k), N-col of B/D
    const int half  = lane >> 4;         // 0: K=0/1, 1: K=2/3
    const int fbase = wave * 4;          // each wave owns 4 f-tiles
    const int n_begin = blockIdx.x * CHUNK;

    v8f accS[4], accH[4];
    for (int i = 0; i < 4; ++i)
        for (int j = 0; j < 8; ++j) { accS[i][j] = 0.0f; accH[i][j] = 0.0f; }

    for (int s = 0; s < CHUNK / 4; ++s) {
        const int n0 = n_begin + 4 * s;
        if (n0 >= N) break;                       // uniform per block
        const int nA = n0 + half * 2;             // A/B VGPR.x sample
        const int nB = nA + 1;                    // A/B VGPR.y sample
        const float vA = (nA < N) ? 1.0f : 0.0f;  // validity masks (no divergence)
        const float vB = (nB < N) ? 1.0f : 0.0f;
        const int cA = (nA < N) ? nA : (N - 1);
        const int cB = (nB < N) ? nB : (N - 1);

        // ---- build B (4x16): row K, striped across lanes ----
        float yA = (col < NCLS) ? tp[cA * NCLS + col] * vA : 0.0f;
        float yB = (col < NCLS) ? tp[cB * NCLS + col] * vB : 0.0f;
        const float qA = row16_sum(yA * yA);      // ||y||^2, broadcast to row
        const float qB = row16_sum(yB * yB);
        v2f b;
        b.x = (col < NCLS) ? yA : (col == 10 ? vA : (col == 11 ? qA : 0.0f));
        b.y = (col < NCLS) ? yB : (col == 10 ? vB : (col == 11 ? qB : 0.0f));

        for (int ff = 0; ff < 4; ++ff) {
            const int f = fbase + ff;
            const int off = f * NBIN + col;       // fk index of this lane's A row
            // ---- A (16x4 fp32): VGPR0 = K0(lo)/K2(hi), VGPR1 = K1/K3 ----
            float ax = mem[(size_t)cA * FKDIM + off] * vA;
            float ay = mem[(size_t)cB * FKDIM + off] * vB;

            // ---- argmax over k: DPP max butterfly + ballot/ctz first-max ----
            const float mx = row16_max(ax);
            const float my = row16_max(ay);
            const unsigned int balx = (unsigned int)__ballot(ax == mx);
            const unsigned int baly = (unsigned int)__ballot(ay == my);
            const unsigned int gx = (balx >> (half << 4)) & 0xFFFFu;
            const unsigned int gy = (baly >> (half << 4)) & 0xFFFFu;
            const int ix = __builtin_ctz(gx | 0x10000u);   // first lane holding max
            const int iy = __builtin_ctz(gy | 0x10000u);

            v2f a;  a.x = ax;  a.y = ay;
            v2f oh; oh.x = (col == ix) ? vA : 0.0f;
                    oh.y = (col == iy) ? vB : 0.0f;

            accS[ff] = wmma_f32x4(a,  b, accS[ff]);
            accH[ff] = wmma_f32x4(oh, b, accH[ff]);
        }
    }

    // ---- accumulate D tiles into global scratch (native f32 atomics) ----
    for (int ff = 0; ff < 4; ++ff) {
        const int f = fbase + ff;
        float* pS = ws + (size_t)(f * 8) * 32 + lane;
        float* pH = pS + NFEAT * 8 * 32;
        for (int v = 0; v < 8; ++v) {
            unsafeAtomicAdd(pS + v * 32, accS[ff][v]);
            unsafeAtomicAdd(pH + v * 32, accH[ff][v]);
        }
    }
}

// D element (f, k, c) lives at (f*8 + (k&7))*32 + ((k>>3)<<4) + c
static __device__ __forceinline__ int widx(int f, int k, int c) {
    return (f * 8 + (k & 7)) * 32 + ((k >> 3) << 4) + c;
}

__global__ __launch_bounds__(512) void vb_final(const float* __restrict__ ws,
                                                float* __restrict__ out, int N) {
    __shared__ float s_cent [NFEAT][NBIN][NCLS];
    __shared__ float s_centH[NFEAT][NBIN][NCLS];
    __shared__ float s_counts[NFEAT][NBIN];
    __shared__ float s_acc[5];   // intra, inter, binvar, valid, cvar
    const int t = threadIdx.x;
    if (t < 5) s_acc[t] = 0.0f;
    __syncthreads();

    const int f = t >> 4, k = t & 15;
    const float* pS = ws;
    const float* pH = ws + NFEAT * 8 * 32;

    // ---- soft losses ----
    float S[NCLS];
    for (int c = 0; c < NCLS; ++c) S[c] = pS[widx(f, k, c)];
    const float massRaw = pS[widx(f, k, 10)];
    const float mys     = pS[widx(f, k, 11)];      // sum_n m * ||y||^2
    const float bm = massRaw + 1e-8f;
    float csq = 0.0f, t3 = 0.0f;
    for (int c = 0; c < NCLS; ++c) {
        const float ce = S[c] / bm;
        s_cent[f][k][c] = ce;
        csq += ce * ce;
        t3  += ce * S[c];
    }
    atomicAdd(&s_acc[0], mys + csq * massRaw - 2.0f * t3);

    // ---- hard per-bin stats ----
    const float counts = pH[widx(f, k, 10)];
    const float sumsq  = pH[widx(f, k, 11)];
    const float safe = fmaxf(counts, 1.0f);
    float chsq = 0.0f;
    for (int c = 0; c < NCLS; ++c) {
        const float ce = pH[widx(f, k, c)] / safe;
        s_centH[f][k][c] = ce;
        chsq += ce * ce;
    }
    s_counts[f][k] = counts;
    atomicAdd(&s_acc[2], (counts > 1.0f) ? (sumsq / safe - chsq) : 0.0f);
    atomicAdd(&s_acc[3], (counts > 1.0f) ? 1.0f : 0.0f);
    __syncthreads();

    if (k < NBIN - 1) {
        float d2 = 0.0f;
        for (int c = 0; c < NCLS; ++c) {
            const float d = s_cent[f][k + 1][c] - s_cent[f][k][c];
            d2 += d * d;
        }
        atomicAdd(&s_acc[1], d2);
    }
    if (k == 0) {  // per-feature hard inter variance
        float M = 0.0f, mean[NCLS];
        for (int c = 0; c < NCLS; ++c) mean[c] = 0.0f;
        for (int kk = 0; kk < NBIN; ++kk) {
            const float m1 = (s_counts[f][kk] >= 1.0f) ? 1.0f : 0.0f;
            M += m1;
            for (int c = 0; c < NCLS; ++c) mean[c] += s_centH[f][kk][c] * m1;
        }
        const float safeM = fmaxf(M, 1.0f);
        for (int c = 0; c < NCLS; ++c) mean[c] /= safeM;
        float cvar = 0.0f;
        for (int kk = 0; kk < NBIN; ++kk) {
            const float m1 = (s_counts[f][kk] >= 1.0f) ? 1.0f : 0.0f;
            for (int c = 0; c < NCLS; ++c) {
                const float d = (s_centH[f][kk][c] - mean[c]) * m1;
                cvar += d * d;
            }
        }
        cvar /= safeM;
        if (!(M > 1.0f)) cvar = 0.0f;
        atomicAdd(&s_acc[4], cvar);
    }
    __syncthreads();

    if (t == 0) {
        const float loss_intra = s_acc[0] / (float)N;
        const float loss_inter = -s_acc[1] / (float)NFEAT;
        const float intra_var  = s_acc[2] / (s_acc[3] + 1e-8f);
        const float inter_var  = s_acc[4] / (float)NFEAT;
        out[0] = loss_intra + 3.0f * loss_inter;
        out[1] = loss_intra;
        out[2] = loss_inter;
        out[3] = intra_var;
        out[4] = inter_var;
    }
}

extern "C" void kernel_launch(void* const* d_in, const int* in_sizes, int n_in,
                              void* d_out, int out_size, void* d_ws, size_t ws_size,
                              hipStream_t stream) {
    (void)n_in; (void)out_size; (void)ws_size;
    const float* membership = (const float*)d_in[0];   // (N, 32, 16) f32
    const float* teacher    = (const float*)d_in[1];   // (N, 10)     f32
    float* out = (float*)d_out;                        // 5 scalars
    float* ws  = (float*)d_ws;                         // >= 64 KB scratch
    const int N = in_sizes[0] / FKDIM;

    vb_zero<<<(WSFLOATS + 255) / 256, 256, 0, stream>>>(ws, WSFLOATS);
    const int nblocks = (N + CHUNK - 1) / CHUNK;
    vb_gemm<<<nblocks, 256, 0, stream>>>(membership, teacher, ws, N);
    vb_final<<<1, 512, 0, stream>>>(ws, out, N);
}